// YOLO_24979529794010
// MI455X (gfx1250) — compile-verified
//
#include <hip/hip_runtime.h>

#define NUM_B     16
#define N_ANCH    16128      // 64*64*3 + 32*32*3 + 16*16*3
#define N_L0      12288
#define N_L01     15360
#define MAX_DET_N 100
#define NCLS      80
#define SCORE_THR 0.25f
#define IOU_THR   0.5f
#define IMG_SZ    512.0f
#define NMS_T     1024       // 32 wave32s: minimizes per-round serial work
#define NWAVES    (NMS_T / 32)

__constant__ float c_anch[9][2] = {
    {12.f, 16.f},  {19.f, 36.f},   {40.f, 28.f},
    {36.f, 75.f},  {76.f, 55.f},   {72.f, 146.f},
    {142.f, 110.f},{192.f, 243.f}, {459.f, 401.f}};

__device__ __forceinline__ float sigmoidf(float x) {
    return 1.0f / (1.0f + __expf(-x));
}

// ---------------------------------------------------------------------------
// Stage 1: streaming decode. One thread per anchor. ~88MB read once -> HBM
// bound (~4us at 23.3 TB/s); compute negligible.
// ---------------------------------------------------------------------------
__global__ __launch_bounds__(256)
void yolo_decode(const float* __restrict__ p0, const float* __restrict__ p1,
                 const float* __restrict__ p2, float* __restrict__ boxes,
                 float* __restrict__ scores, float* __restrict__ cls)
{
    int tid = blockIdx.x * blockDim.x + threadIdx.x;
    if (tid >= NUM_B * N_ANCH) return;
    int b = tid / N_ANCH;
    int r = tid - b * N_ANCH;

    const float* p; int off, S, abase; float stride;
    if (r < N_L0)       { p = p0; off = r;         S = 64; stride = 8.f;  abase = 0; }
    else if (r < N_L01) { p = p1; off = r - N_L0;  S = 32; stride = 16.f; abase = 3; }
    else                { p = p2; off = r - N_L01; S = 16; stride = 32.f; abase = 6; }

    const float* q = p + ((size_t)b * (size_t)(S * S * 3) + (size_t)off) * 85;

    int a    = off % 3;
    int cell = off / 3;
    int gx   = cell % S;      // meshgrid 'xy': grid[i,j] = (j, i)
    int gy   = cell / S;

    float bx = (sigmoidf(q[0]) * 2.f - 0.5f + (float)gx) * stride;
    float by = (sigmoidf(q[1]) * 2.f - 0.5f + (float)gy) * stride;
    float sw = sigmoidf(q[2]) * 2.f;
    float sh = sigmoidf(q[3]) * 2.f;
    float bw = sw * sw * c_anch[abase + a][0];   // wh * (anchors_px/stride) * stride
    float bh = sh * sh * c_anch[abase + a][1];

    // sigmoid is monotonic: argmax/max over logits == over sigmoid(logits)
    float m = q[5]; int mi = 0;
    for (int c = 1; c < NCLS; ++c) {
        float v = q[5 + c];
        if (v > m) { m = v; mi = c; }          // strict '>' keeps first index
    }
    float score = sigmoidf(q[4]) * sigmoidf(m);

    bx = fminf(fmaxf(bx, 0.f), IMG_SZ);
    by = fminf(fmaxf(by, 0.f), IMG_SZ);
    bw = fminf(fmaxf(bw, 0.f), IMG_SZ);
    bh = fminf(fmaxf(bh, 0.f), IMG_SZ);

    size_t o = (size_t)b * N_ANCH + r;
    reinterpret_cast<float4*>(boxes)[o] = make_float4(bx, by, bw, bh);
    scores[o] = (score >= SCORE_THR) ? score : -1.0f;   // threshold sentinel
    cls[o]    = (float)mi;
}

// ---------------------------------------------------------------------------
// Stage 2: greedy DIoU-NMS, one workgroup (1024 threads = 32 wave32s) per
// image. 100 sequential rounds of argmax+suppress: latency bound, so scores
// live in LDS for all rounds (63KB << 320KB WGP LDS) and boxes stay
// L2-resident (6MB << 192MB L2). Score staging uses the CDNA5 async
// global->LDS DMA path (ASYNCcnt + s_wait_asynccnt).
// ---------------------------------------------------------------------------
__global__ __launch_bounds__(NMS_T)
void yolo_nms(const float* __restrict__ boxes, const float* __restrict__ scores,
              const float* __restrict__ cls, float* __restrict__ out)
{
    __shared__ float s_sc[N_ANCH];        // 64512 B
    __shared__ float s_wv[NWAVES];        // per-wave argmax value
    __shared__ int   s_wi[NWAVES];        // per-wave argmax index
    __shared__ float s_bv;
    __shared__ int   s_bi;

    const int b = blockIdx.x;
    const int t = threadIdx.x;
    const float*  gsc = scores + (size_t)b * N_ANCH;
    const float4* gbx = reinterpret_cast<const float4*>(boxes) + (size_t)b * N_ANCH;
    const float*  gcl = cls + (size_t)b * N_ANCH;

    // ---- stage score vector into LDS via async-to-LDS b128 copies ----
#if defined(__gfx1250__)
    {
        unsigned lds_base = (unsigned)(size_t)(const void*)&s_sc[0]; // low 32b = LDS offset
        for (int c = t; c < N_ANCH / 4; c += NMS_T) {
            unsigned lo = lds_base + (unsigned)c * 16u;
            unsigned go = (unsigned)c * 16u;
            asm volatile("global_load_async_to_lds_b128 %0, %1, %2 offset:0"
                         :: "v"(lo), "v"(go), "s"(gsc) : "memory");
        }
        asm volatile("s_wait_asynccnt 0" ::: "memory");
    }
#else
    for (int j = t; j < N_ANCH; j += NMS_T) s_sc[j] = gsc[j];
#endif
    __syncthreads();

    for (int it = 0; it < MAX_DET_N; ++it) {
        // ---- argmax over LDS scores (first-index tie-break, jnp.argmax) ----
        float mv = -3.0e38f;
        int   mi = 0;
        for (int j = t; j < N_ANCH; j += NMS_T) {
            float v = s_sc[j];
            if (v > mv) { mv = v; mi = j; }   // strided: indices increase per thread
        }
        for (int off = 16; off > 0; off >>= 1) {         // wave32 shuffle reduce
            float ov = __shfl_down(mv, off, 32);
            int   oi = __shfl_down(mi, off, 32);
            if (ov > mv || (ov == mv && oi < mi)) { mv = ov; mi = oi; }
        }
        if ((t & 31) == 0) { s_wv[t >> 5] = mv; s_wi[t >> 5] = mi; }
        __syncthreads();
        if (t == 0) {
            float bvv = s_wv[0]; int bii = s_wi[0];
            for (int w = 1; w < NWAVES; ++w) {
                float v = s_wv[w]; int i = s_wi[w];
                if (v > bvv || (v == bvv && i < bii)) { bvv = v; bii = i; }
            }
            s_bv = bvv; s_bi = bii;
        }
        __syncthreads();
        const float bs = s_bv;
        const int   bi = s_bi;
        const float4 bb = gbx[bi];           // broadcast L2 hit

        if (t == 0) {
            float* o = out + ((size_t)b * MAX_DET_N + it) * 6;
            const bool valid = bs > 0.0f;
            o[0] = valid ? bb.x : 0.f;
            o[1] = valid ? bb.y : 0.f;
            o[2] = valid ? bb.z : 0.f;
            o[3] = valid ? bb.w : 0.f;
            o[4] = valid ? bs   : 0.f;
            o[5] = valid ? gcl[bi] : 0.f;
        }

        // ---- DIoU suppression against all anchors ----
        const float b1nx = bb.x - bb.z * 0.5f, b1ny = bb.y - bb.w * 0.5f;
        const float b1xx = bb.x + bb.z * 0.5f, b1xy = bb.y + bb.w * 0.5f;
        const float area1 = bb.z * bb.w;

        for (int j = t; j < N_ANCH; j += NMS_T) {
            float4 qb = gbx[j];
            float b2nx = qb.x - qb.z * 0.5f, b2ny = qb.y - qb.w * 0.5f;
            float b2xx = qb.x + qb.z * 0.5f, b2xy = qb.y + qb.w * 0.5f;
            float iw = fmaxf(fminf(b1xx, b2xx) - fmaxf(b1nx, b2nx), 0.f);
            float ih = fmaxf(fminf(b1xy, b2xy) - fmaxf(b1ny, b2ny), 0.f);
            float inter = iw * ih;
            float uni   = area1 + qb.z * qb.w - inter;
            float iou   = inter / (uni + 1e-9f);
            float ex = fmaxf(b1xx, b2xx) - fminf(b1nx, b2nx);
            float ey = fmaxf(b1xy, b2xy) - fminf(b1ny, b2ny);
            float c2 = ex * ex + ey * ey + 1e-9f;
            float dx = bb.x - qb.x, dy = bb.y - qb.y;
            float diou = iou - (dx * dx + dy * dy) / c2;
            // reference: suppress ious>TH AND force .at[i].set(-1)
            if (diou > IOU_THR || j == bi) s_sc[j] = -1.0f;
        }
        __syncthreads();
    }
}

extern "C" void kernel_launch(void* const* d_in, const int* in_sizes, int n_in,
                              void* d_out, int out_size, void* d_ws, size_t ws_size,
                              hipStream_t stream) {
    (void)in_sizes; (void)n_in; (void)out_size; (void)ws_size;
    const float* p0 = (const float*)d_in[0];   // [16,64,64,3,85]
    const float* p1 = (const float*)d_in[1];   // [16,32,32,3,85]
    const float* p2 = (const float*)d_in[2];   // [16,16,16,3,85]
    float* out = (float*)d_out;                // [16,100,6]

    float* boxes  = (float*)d_ws;                               // 16*16128*4 f32
    float* scores = boxes  + (size_t)NUM_B * N_ANCH * 4;        // 16*16128 f32
    float* clsbuf = scores + (size_t)NUM_B * N_ANCH;            // 16*16128 f32

    const int total = NUM_B * N_ANCH;
    yolo_decode<<<(total + 255) / 256, 256, 0, stream>>>(p0, p1, p2,
                                                        boxes, scores, clsbuf);
    yolo_nms<<<NUM_B, NMS_T, 0, stream>>>(boxes, scores, clsbuf, out);
}